// TreeBasedSpatialEmbedding_40707700032516
// MI455X (gfx1250) — compile-verified
//
#include <hip/hip_runtime.h>
#include <cstdint>
#include <cstddef>

// out[b, d, n, 0] = embed_node[n, d]   (f32 transpose + batch broadcast)
// N=16384, D=128, B=32. 8 MB read, 256 MB write -> pure store-BW problem.

#define N_NODES 16384
#define EMBED_D 128
#define BATCH   32
#define TN      128   // n-rows per block tile
#define BPB     8     // batches per block
#define THREADS 256   // 8 wave32s

typedef __attribute__((ext_vector_type(4))) float v4f;

__global__ __launch_bounds__(THREADS) void transpose_bcast_kernel(
    const float* __restrict__ emb, float* __restrict__ out)
{
    // 64 KB tile, stored row-major in 16B chunks with XOR swizzle:
    //   float index (r, d) = r*128 + (dc ^ ((r>>2)&31))*4 + (d&3),  dc = d>>2
    __shared__ float tile[TN * EMBED_D];

    const int tid  = threadIdx.x;
    const int lane = tid & 31;
    const int wave = tid >> 5;
    const int n0   = blockIdx.x * TN;
    const int b0   = blockIdx.y * BPB;

    // Flat->LDS: hardware uses addr[31:0] as the LDS byte address, so the low
    // 32 bits of the generic pointer are the wave-logical LDS offset.
    const uint32_t lds_base = (uint32_t)(uintptr_t)(&tile[0]);

    // ---- Phase 1: async DMA global -> LDS (coalesced 512 B per wave-instr) ----
#pragma unroll
    for (int i = 0; i < TN / 8; ++i) {
        const int r   = i * 8 + wave;              // n within tile
        const int dc  = lane;                      // d-chunk (4 floats)
        const int dcs = dc ^ ((r >> 2) & 31);      // swizzled chunk
        const uint64_t gaddr =
            (uint64_t)(uintptr_t)(emb + (size_t)(n0 + r) * EMBED_D + (size_t)dc * 4);
        const uint32_t laddr =
            lds_base + (uint32_t)((r * EMBED_D + dcs * 4) * sizeof(float));
        asm volatile("global_load_async_to_lds_b128 %0, %1, off"
                     :: "v"(laddr), "v"(gaddr)
                     : "memory");
    }
#if __has_builtin(__builtin_amdgcn_s_wait_asynccnt)
    __builtin_amdgcn_s_wait_asynccnt(0);
#else
    asm volatile("s_wait_asynccnt 0" ::: "memory");
#endif
    __syncthreads();

    // ---- Phase 2: transpose-on-read, replicate across 8 batches, NT stores ----
#pragma unroll
    for (int p = 0; p < EMBED_D / 8; ++p) {        // 16 passes, wave owns one d
        const int d   = p * 8 + wave;
        const int dc  = d >> 2;
        const int dl  = d & 3;
        const int dcs = dc ^ lane;                  // (r>>2)&31 == lane for r=4*lane+i

        const float* trow = &tile[(4 * lane) * EMBED_D + dcs * 4 + dl];
        v4f v;
        v.x = trow[0 * EMBED_D];
        v.y = trow[1 * EMBED_D];
        v.z = trow[2 * EMBED_D];
        v.w = trow[3 * EMBED_D];

        // out flat index: ((b*128 + d) * 16384) + n0 + 4*lane   (16B aligned)
        size_t base = ((size_t)b0 * EMBED_D + (size_t)d) * (size_t)N_NODES
                      + (size_t)n0 + (size_t)(4 * lane);
#pragma unroll
        for (int b = 0; b < BPB; ++b) {
            __builtin_nontemporal_store(
                v, (v4f*)(out + base + (size_t)b * (size_t)EMBED_D * (size_t)N_NODES));
        }
    }
}

extern "C" void kernel_launch(void* const* d_in, const int* in_sizes, int n_in,
                              void* d_out, int out_size, void* d_ws, size_t ws_size,
                              hipStream_t stream) {
    const float* emb = (const float*)d_in[0];   // embed_node [16384, 128] f32
    float* out = (float*)d_out;                 // [32, 128, 16384, 1] f32

    dim3 grid(N_NODES / TN, BATCH / BPB);       // (128, 4)
    transpose_bcast_kernel<<<grid, THREADS, 0, stream>>>(emb, out);
}